// FAModule_78838419685958
// MI455X (gfx1250) — compile-verified
//
#include <hip/hip_runtime.h>
#include <hip/hip_bf16.h>

typedef float v2f __attribute__((ext_vector_type(2)));
typedef float v8f __attribute__((ext_vector_type(8)));

// ---------------------------------------------------------------------------
// Pre-kernel: min over design_matrix (448 elems) -> min_diffusivity scalar
// ---------------------------------------------------------------------------
__global__ void fa_md_kernel(const float* __restrict__ dm, int n, float* __restrict__ md_out) {
    __shared__ float red[256];
    int t = threadIdx.x;
    float m = 3.4e38f;
    for (int i = t; i < n; i += 256) m = fminf(m, dm[i]);
    red[t] = m;
    __syncthreads();
    for (int s = 128; s > 0; s >>= 1) {
        if (t < s) red[t] = fminf(red[t], red[t + s]);
        __syncthreads();
    }
    if (t == 0) md_out[0] = 1e-6f / (-red[0]);   // TOL / (-min(design_matrix))
}

// ---------------------------------------------------------------------------
// Main kernel: log-linear fit via V_WMMA_F32_16X16X4_F32 + analytic eigensolve
// 256 threads = 8 waves; each wave handles 32 voxels (two 16x16 D tiles).
//
// Trick: ln(x) = ln2 * log2(x); ln2 is folded into the (tiny) B matrix, so the
// per-element path is just v_max_f32 + native v_log_f32 (inputs are clipped to
// md ~ 5e-10, so no denorm/inf fixups are needed -> raw v_log_f32 is exact
// enough and ~4x fewer VALU ops than the OCML logf expansion).
// ---------------------------------------------------------------------------
__global__ __launch_bounds__(256) void fa_main_kernel(
    const float* __restrict__ data,      // [nvox][64]
    const float* __restrict__ dm_inv,    // [7][64]
    const float* __restrict__ md_ptr,    // scalar min_diffusivity
    float* __restrict__ out,             // [nvox]
    int nvox)
{
    __shared__ float sW[16 * 64];          // padded pinv * ln2: coeffs 7..15 = 0
    __shared__ float sStage[8][32 * 17];   // per-wave D staging, pad 17 vs banks

    const int tid  = threadIdx.x;
    const int wave = tid >> 5;
    const int lane = tid & 31;
    const int row  = lane & 15;   // A-row within tile / B coeff column
    const int hi   = lane >> 4;   // lane half selects K offset {0,2}

    // stage padded pinv (scaled by ln2) into LDS
    const float LN2 = 0.69314718055994531f;
    for (int i = tid; i < 16 * 64; i += 256) {
        int c = i >> 6, g = i & 63;
        sW[i] = (c < 7) ? (LN2 * dm_inv[c * 64 + g]) : 0.0f;
    }
    const float md = md_ptr[0];
    __syncthreads();

    const int waveBase = blockIdx.x * 256 + wave * 32;

    // clamp load rows so EXEC stays all-ones even on a ragged tail
    const int r0 = min(waveBase + row,      nvox - 1);
    const int r1 = min(waveBase + 16 + row, nvox - 1);
    const float* __restrict__ arow0 = data + (long long)r0 * 64;
    const float* __restrict__ arow1 = data + (long long)r1 * 64;

    v8f acc0 = {};   // voxels 0..15 of wave  x  coeffs 0..15
    v8f acc1 = {};   // voxels 16..31 of wave x  coeffs 0..15

    #pragma unroll
    for (int kk = 0; kk < 16; ++kk) {
        const int g0 = kk * 4 + hi * 2;    // this lane's K pair within K=64

        // B operand: B[k][n] = ln2 * pinv[n][g0 + j]   (j = vgpr index)
        v2f b;
        b.x = sW[row * 64 + g0];
        b.y = sW[row * 64 + g0 + 1];

        // A operands: clip + native log2 (ln2 folded into B)
        const float2 d0 = *(const float2*)(arow0 + g0);
        const float2 d1 = *(const float2*)(arow1 + g0);
        v2f a0, a1;
        a0.x = __builtin_amdgcn_logf(fmaxf(d0.x, md));
        a0.y = __builtin_amdgcn_logf(fmaxf(d0.y, md));
        a1.x = __builtin_amdgcn_logf(fmaxf(d1.x, md));
        a1.y = __builtin_amdgcn_logf(fmaxf(d1.y, md));

        acc0 = __builtin_amdgcn_wmma_f32_16x16x4_f32(false, a0, false, b,
                                                     (short)0, acc0, false, false);
        acc1 = __builtin_amdgcn_wmma_f32_16x16x4_f32(false, a1, false, b,
                                                     (short)0, acc1, false, false);
    }

    // Scatter D tiles to LDS: element (M=i or 8+i, N=row) per ISA C/D layout
    float* st = sStage[wave];
    #pragma unroll
    for (int i = 0; i < 8; ++i) {
        const int v = (hi ? 8 + i : i);
        st[v * 17 + row]        = acc0[i];
        st[(16 + v) * 17 + row] = acc1[i];
    }
    __syncthreads();

    // ---- per-lane: one voxel's symmetric 3x3 eigensolve + FA ----
    const float* fv = &st[lane * 17];
    const float dxx = fv[0], dxy = fv[1], dyy = fv[2];
    const float dxz = fv[3], dyz = fv[4], dzz = fv[5];

    const float q  = (dxx + dyy + dzz) * (1.0f / 3.0f);
    const float p1 = dxy * dxy + dxz * dxz + dyz * dyz;
    const float bxx = dxx - q, byy = dyy - q, bzz = dzz - q;
    const float p2 = bxx * bxx + byy * byy + bzz * bzz + 2.0f * p1;
    const float p  = sqrtf(p2 * (1.0f / 6.0f));
    const float pinv = (p > 1e-30f) ? (1.0f / p) : 0.0f;   // branchless iso guard

    const float detB = bxx * (byy * bzz - dyz * dyz)
                     - dxy * (dxy * bzz - dyz * dxz)
                     + dxz * (dxy * dyz - byy * dxz);
    float r = 0.5f * detB * pinv * pinv * pinv;
    r = fminf(1.0f, fmaxf(-1.0f, r));
    const float phi = acosf(r) * (1.0f / 3.0f);

    float e1 = q + 2.0f * p * __cosf(phi);
    float e3 = q + 2.0f * p * __cosf(phi + 2.09439510239319549f); // +2pi/3
    float e2 = 3.0f * q - e1 - e3;

    // reference semantics: clip to md, NaN -> 0
    e1 = (e1 != e1) ? 0.0f : fmaxf(e1, md);
    e2 = (e2 != e2) ? 0.0f : fmaxf(e2, md);
    e3 = (e3 != e3) ? 0.0f : fmaxf(e3, md);

    const float allz = ((e1 == 0.0f) && (e2 == 0.0f) && (e3 == 0.0f)) ? 1.0f : 0.0f;
    const float num = 0.5f * ((e1 - e2) * (e1 - e2) +
                              (e2 - e3) * (e2 - e3) +
                              (e3 - e1) * (e3 - e1));
    const float den = e1 * e1 + e2 * e2 + e3 * e3 + allz;
    const float fa  = sqrtf(num / den);

    const int vox = waveBase + lane;
    if (vox < nvox) out[vox] = fa;
}

// ---------------------------------------------------------------------------
extern "C" void kernel_launch(void* const* d_in, const int* in_sizes, int n_in,
                              void* d_out, int out_size, void* d_ws, size_t ws_size,
                              hipStream_t stream) {
    const float* data   = (const float*)d_in[0];   // [1,80,80,48,64]
    const float* dm     = (const float*)d_in[1];   // [64,7]
    const float* dm_inv = (const float*)d_in[2];   // [7,64]
    float* out = (float*)d_out;
    float* md  = (float*)d_ws;

    const int nvox = in_sizes[0] / 64;             // 307200

    fa_md_kernel<<<1, 256, 0, stream>>>(dm, in_sizes[1], md);

    const int nblocks = (nvox + 255) / 256;        // 1200
    fa_main_kernel<<<nblocks, 256, 0, stream>>>(data, dm_inv, md, out, nvox);
}